// PhysicsLayer_60009283060132
// MI455X (gfx1250) — compile-verified
//
#include <hip/hip_runtime.h>
#include <stdint.h>
#include <math.h>

#define B_LEN 32
#define T_LEN 4096
#define NCH   15
#define NF    14
#define BLOCK 256
#define HALO  51                   // PE needs 51 back; hurst needs 15; ret needs 1
#define TILE_ROWS (BLOCK + HALO)   // 307

#if defined(__gfx1250__) && __has_builtin(__builtin_amdgcn_tensor_load_to_lds) && __has_builtin(__builtin_amdgcn_s_wait_tensorcnt)
#define USE_TDM 1
typedef __attribute__((ext_vector_type(4))) unsigned int tdm_v4u;
typedef __attribute__((ext_vector_type(8))) int          tdm_v8i;
typedef __attribute__((ext_vector_type(4))) int          tdm_v4i;
#else
#define USE_TDM 0
#endif

// ---------------------------------------------------------------------------
// Kernel A: all time-parallel features. One block = 256 consecutive t for one
// batch row. The 307x15 f32 input tile (strided rows) is staged into LDS via
// the Tensor Data Mover (one DMA per block), then everything reads LDS.
// Writes psi channels 0,1,4..13. Channels 2,3 come from the EKF kernel.
// ---------------------------------------------------------------------------
__global__ __launch_bounds__(BLOCK) void physics_features_kernel(
    const float* __restrict__ x, float* __restrict__ out) {
  __shared__ __align__(16) float s_x[TILE_ROWS * NCH];
  __shared__ float s_logc[TILE_ROWS];

  const int tid = threadIdx.x;
  const int b   = blockIdx.y;
  const int t0  = blockIdx.x * BLOCK;
  const float* xb = x + (size_t)b * T_LEN * NCH;

#if USE_TDM
  {
    const bool first    = (t0 == 0);
    const int halo_rows = first ? 0 : HALO;
    const int nrows     = BLOCK + halo_rows;
    if (tid < 32) {  // wave 0 issues the DMA (TDM ignores EXEC; one issue per wave)
      const uint64_t ga  = (uint64_t)(uintptr_t)(xb + (size_t)(t0 - halo_rows) * NCH);
      const uint32_t lds = (uint32_t)(uintptr_t)(&s_x[0]) +
                           (first ? (uint32_t)(HALO * NCH * 4) : 0u);
      // D# group 0: count=1 (valid user descriptor), lds_addr, 57b global addr, type=2
      tdm_v4u g0;
      g0[0] = 1u;
      g0[1] = lds;
      g0[2] = (uint32_t)(ga & 0xFFFFFFFFull);
      g0[3] = (uint32_t)((ga >> 32) & 0x01FFFFFFull) | 0x80000000u;
      // D# group 1: data_size=4B; tensor 15 x nrows; tile 15 x nrows; dim0 stride=15
      const uint32_t td0 = NCH, tl0 = NCH;
      const uint32_t td1 = (uint32_t)nrows, tl1 = (uint32_t)nrows;
      const uint64_t st0 = NCH;  // elements
      tdm_v8i g1;
      g1[0] = (int)(2u << 16);                                   // data_size=4B
      g1[1] = (int)((td0 & 0xFFFFu) << 16);                      // tensor_dim0[15:0]
      g1[2] = (int)((td0 >> 16) | ((td1 & 0xFFFFu) << 16));      // td0 hi | td1 lo
      g1[3] = (int)((td1 >> 16) | ((tl0 & 0xFFFFu) << 16));      // td1 hi | tile_dim0
      g1[4] = (int)(tl1 & 0xFFFFu);                              // tile_dim1, tile_dim2=0
      g1[5] = (int)(st0 & 0xFFFFFFFFull);                        // dim0_stride lo
      g1[6] = (int)((st0 >> 32) & 0xFFFFull);                    // dim0_stride hi, d1s=0
      g1[7] = 0;
      tdm_v4i g2 = {0, 0, 0, 0};                                 // 2-D tensor: groups 2/3 unused
      tdm_v4i g3 = {0, 0, 0, 0};
      tdm_v8i g4 = {0, 0, 0, 0, 0, 0, 0, 0};                     // VADDR4: unused/NULL group
      __builtin_amdgcn_tensor_load_to_lds(g0, g1, g2, g3, g4, 0);
      __builtin_amdgcn_s_wait_tensorcnt((short)0);
    }
    __syncthreads();
    if (first) {  // edge-pad: replicate row t=0 into the halo
      for (int i = tid; i < HALO * NCH; i += BLOCK)
        s_x[i] = s_x[HALO * NCH + (i % NCH)];
      __syncthreads();
    }
  }
#else
  for (int i = tid; i < TILE_ROWS * NCH; i += BLOCK) {
    int row = t0 - HALO + i / NCH;
    if (row < 0) row = 0;  // edge padding
    s_x[i] = xb[(size_t)row * NCH + (i % NCH)];
  }
  __syncthreads();
#endif

  for (int i = tid; i < TILE_ROWS; i += BLOCK)
    s_logc[i] = logf(s_x[i * NCH + 3] + 1e-8f);
  __syncthreads();

  const int L = tid + HALO;
  const float* r = &s_x[L * NCH];
  const float open_p = r[0], high_p = r[1], low_p = r[2], close_p = r[3];
  const float ema20 = r[5], rvol = r[6];

  const float lc  = s_logc[L];
  const float ret = lc - s_logc[L - 1];   // t=0: halo replication => 0
  const float vol = fabsf(ret);

  // ---- Hurst proxy: window 16 over log-close; two-pass var, ddof=1 ----
  float mn = s_logc[L - 15], mx = mn, sum = mn;
#pragma unroll
  for (int k = 1; k < 16; ++k) {
    const float w = s_logc[L - 15 + k];
    mn = fminf(mn, w); mx = fmaxf(mx, w); sum += w;
  }
  const float mean = sum * (1.0f / 16.0f);
  float sd2 = 0.0f;
#pragma unroll
  for (int k = 0; k < 16; ++k) {
    const float d = s_logc[L - 15 + k] - mean;
    sd2 += d * d;
  }
  const float R   = (mx - mn) + 1e-8f;
  const float Sd  = sqrtf(sd2 / 15.0f) + 1e-8f;
  const float dev = logf(R / Sd) - 0.5f * logf(16.0f);
  const float h   = 0.5f + 0.3f * tanhf(2.0f * dev);

  // ---- Permutation entropy: order 3, delay 1, window 50 over close ----
  // stable argsort of (a0,a1,a2) -> rank of each element; 6 reachable perms
  // counted in one u64 (6 x 8-bit counters).
  unsigned long long cnt = 0ull;
  float a0 = s_x[(tid + 0) * NCH + 3];
  float a1 = s_x[(tid + 1) * NCH + 3];
  float a2 = s_x[(tid + 2) * NCH + 3];
#pragma unroll 5
  for (int pos = 0; pos < 50; ++pos) {
    const int pa = (int)(a1 < a0) + (int)(a2 < a0);
    const int pb = (int)(a0 <= a1) + (int)(a2 < a1);
    const int pc = (int)(a0 <= a2) + (int)(a1 <= a2);
    const int idx6 = pa * 2 + (int)(pb > pc);
    cnt += 1ull << (idx6 << 3);
    if (pos < 49) { a0 = a1; a1 = a2; a2 = s_x[(tid + pos + 3) * NCH + 3]; }
  }
  float ent = 0.0f;
#pragma unroll
  for (int k = 0; k < 6; ++k) {
    const float p = (float)((cnt >> (k * 8)) & 0xFFull) / 50.0f + 1e-8f;
    ent -= p * logf(p);
  }
  ent += 21.0f * 1.8420681e-07f;            // 21 never-hit ids: -(1e-8*log 1e-8) each
  const float pe = ent / 1.7917594692280550f;  // log(3!)

  // ---- simple features + output ----
  const float ema_div = (close_p - ema20) / (ema20 + 1e-8f);
  const float hl      = (high_p - low_p) / (close_p + 1e-8f);
  const float body    = (close_p - open_p) / (close_p + 1e-8f);

  float* o = out + ((size_t)b * T_LEN + (size_t)(t0 + tid)) * NF;
  o[0]  = h;
  o[1]  = vol;
  o[4]  = pe;
  o[5]  = ret;
  o[6]  = ema_div;
  o[7]  = rvol;
  o[8]  = hl;
  o[9]  = body;
  o[10] = r[7]  - r[8];    // pin_bull - pin_bear
  o[11] = r[10] - r[9];    // sweep_low - sweep_high
  o[12] = r[11] - r[12];   // fvg_bull - fvg_bear
  o[13] = r[13] - r[14];   // mss_bull - mss_bear
}

// ---------------------------------------------------------------------------
// Kernel B: EKF scan. 32 independent serial chains = exactly one wave32
// (lane b = batch b). Reads h/vol from psi[...,0/1] written by kernel A,
// recomputes z = log(close+1e-8). Writes psi[...,2] (ekf_v), psi[...,3] (res).
// global_prefetch_b8 pre-touches rows 16 steps ahead of the serial chain.
// ---------------------------------------------------------------------------
__global__ __launch_bounds__(32) void ekf_kernel(const float* __restrict__ x,
                                                 float* __restrict__ out) {
  const int b = threadIdx.x;
  if (b >= B_LEN) return;
  const float* xb = x + (size_t)b * T_LEN * NCH;
  float* ob = out + (size_t)b * T_LEN * NF;

  float p = logf(xb[3] + 1e-8f);
  float v = logf(xb[NCH + 3] + 1e-8f) - p;
  float P00 = 1.0f, P01 = 0.0f, P10 = 0.0f, P11 = 1.0f;

  for (int t = 0; t < T_LEN; ++t) {
    if (t + 16 < T_LEN) {
      __builtin_prefetch(&xb[(size_t)(t + 16) * NCH + 3], 0, 0);
      __builtin_prefetch(&ob[(size_t)(t + 16) * NF], 0, 0);
    }
    const float z   = logf(xb[(size_t)t * NCH + 3] + 1e-8f);
    const float hh  = ob[(size_t)t * NF + 0];
    const float sig = ob[(size_t)t * NF + 1];

    const float sgm   = 1.0f / (1.0f + expf(-10.0f * (hh - 0.5f)));
    const float a     = 0.5f + 0.5f * sgm;          // dt * rho, dt=1
    const float scale = fmaxf(100.0f * sig, 1.0f);
    const float q     = 0.1f * scale;

    // predict (mirrors einsum evaluation order)
    const float xp   = p + a * v;
    const float Pp01 = P01 + a * P11;
    const float Pp10 = P10 + a * P11;
    const float Pp00 = (P00 + a * P10) + a * Pp01 + q;
    const float Pp11 = P11 + q;

    // update
    const float y    = z - xp;
    const float Sinv = 1.0f / (Pp00 + scale + 1e-6f);
    const float K0   = Pp00 * Sinv;
    const float K1   = Pp10 * Sinv;

    p = xp + K0 * y;
    v = v + K1 * y;
    const float om = 1.0f - K0;
    P00 = om * Pp00;
    P01 = om * Pp01;
    P10 = Pp10 - K1 * Pp00;   // P stays (slightly) asymmetric, as in reference
    P11 = Pp11 - K1 * Pp01;

    ob[(size_t)t * NF + 2] = v;
    ob[(size_t)t * NF + 3] = z - p;
  }
}

extern "C" void kernel_launch(void* const* d_in, const int* in_sizes, int n_in,
                              void* d_out, int out_size, void* d_ws, size_t ws_size,
                              hipStream_t stream) {
  (void)in_sizes; (void)n_in; (void)d_ws; (void)ws_size; (void)out_size;
  const float* x = (const float*)d_in[0];
  float* out = (float*)d_out;

  dim3 grid(T_LEN / BLOCK, B_LEN);  // 16 x 32 blocks
  physics_features_kernel<<<grid, BLOCK, 0, stream>>>(x, out);
  ekf_kernel<<<1, 32, 0, stream>>>(x, out);
}